// SpatialAttention_83064667504915
// MI455X (gfx1250) — compile-verified
//
#include <hip/hip_runtime.h>
#include <math.h>

typedef __attribute__((ext_vector_type(2))) float v2f;
typedef __attribute__((ext_vector_type(4))) float v4f;
typedef __attribute__((ext_vector_type(8))) float v8f;

#define BATCH 32
#define CH    128
#define HH    128
#define WW    128
#define SPATIAL (HH*WW)        /* 16384 */
#define NPOS (BATCH*SPATIAL)   /* 524288 */

// ---------------- workspace layout (bytes) ----------------
// 64     : float params[8]  {scale0,bias0,scale1,bias1,scale2,bias2}
// 1024   : double bp1[512*4]  per-block BN1 partials {sum_avg,sq_avg,sum_max,sq_max}
// 17408  : double bp2[256*2]  per-block BN2 partials {sum,sq}
// 32768  : float feat0[NPOS]   (channel-avg map)
// +4*NPOS: float feat1[NPOS]   (channel-max map)
// +8*NPOS: float convo[NPOS]   (conv output)

// ============ Kernel 1: channel mean + max, BN1 partial stats ============
__global__ void k_reduce(const float* __restrict__ x,
                         float* __restrict__ feat0, float* __restrict__ feat1,
                         double* __restrict__ bp1) {
  int gid = blockIdx.x * 256 + threadIdx.x;          // [0, NPOS/4)
  int b   = gid / (SPATIAL/4);
  int hw4 = (gid % (SPATIAL/4)) * 4;
  const v4f* xp = (const v4f*)(x + (size_t)b*CH*SPATIAL + hw4);

  v4f s = {0.f,0.f,0.f,0.f};
  v4f m = {-__builtin_inff(),-__builtin_inff(),-__builtin_inff(),-__builtin_inff()};
  #pragma unroll 8
  for (int c = 0; c < CH; ++c) {
    v4f v = xp[c * (SPATIAL/4)];
    s.x += v.x; s.y += v.y; s.z += v.z; s.w += v.w;
    m.x = fmaxf(m.x, v.x); m.y = fmaxf(m.y, v.y);
    m.z = fmaxf(m.z, v.z); m.w = fmaxf(m.w, v.w);
  }
  const float inv = 1.0f / CH;
  v4f a = {s.x*inv, s.y*inv, s.z*inv, s.w*inv};
  *(v4f*)(feat0 + (size_t)b*SPATIAL + hw4) = a;
  *(v4f*)(feat1 + (size_t)b*SPATIAL + hw4) = m;

  double q[4];
  q[0] = (double)a.x + a.y + a.z + a.w;
  q[1] = (double)a.x*a.x + (double)a.y*a.y + (double)a.z*a.z + (double)a.w*a.w;
  q[2] = (double)m.x + m.y + m.z + m.w;
  q[3] = (double)m.x*m.x + (double)m.y*m.y + (double)m.z*m.z + (double)m.w*m.w;

  __shared__ double sh[8][4];
  int lane = threadIdx.x & 31, wave = threadIdx.x >> 5;
  #pragma unroll
  for (int k = 0; k < 4; ++k)
    for (int off = 16; off > 0; off >>= 1)
      q[k] += __shfl_down(q[k], off, 32);
  if (lane == 0) { sh[wave][0]=q[0]; sh[wave][1]=q[1]; sh[wave][2]=q[2]; sh[wave][3]=q[3]; }
  __syncthreads();
  if (threadIdx.x == 0) {
    double r0=0, r1=0, r2=0, r3=0;
    for (int w = 0; w < 8; ++w) { r0+=sh[w][0]; r1+=sh[w][1]; r2+=sh[w][2]; r3+=sh[w][3]; }
    bp1[blockIdx.x*4+0]=r0; bp1[blockIdx.x*4+1]=r1;
    bp1[blockIdx.x*4+2]=r2; bp1[blockIdx.x*4+3]=r3;
  }
}

// ============ Kernel 2: finalize BN1 (deterministic reduce of 512 partials) ============
__global__ void k_bn1(const double* __restrict__ bp1,
                      const float* __restrict__ gamma, const float* __restrict__ beta,
                      float* __restrict__ params) {
  __shared__ double sred[256];
  __shared__ double res[4];
  int t = threadIdx.x;
  for (int q = 0; q < 4; ++q) {
    sred[t] = bp1[t*4+q] + bp1[(t+256)*4+q];
    __syncthreads();
    for (int s = 128; s > 0; s >>= 1) { if (t < s) sred[t] += sred[t+s]; __syncthreads(); }
    if (t == 0) res[q] = sred[0];
    __syncthreads();
  }
  if (t == 0) {
    for (int c = 0; c < 2; ++c) {
      double mean = res[2*c]   / (double)NPOS;
      double var  = res[2*c+1] / (double)NPOS - mean*mean;
      double sc   = (double)gamma[c] / sqrt(var + 1e-5);
      params[2*c]   = (float)sc;
      params[2*c+1] = (float)((double)beta[c] - mean*sc);
    }
  }
}

// ============ Kernel 3: 7x7 conv via V_WMMA_F32_16X16X4_F32 + BN2 partials ============
// Block = 256 thr (8 waves) handles one 16-row x 128-col strip of one image.
// Toeplitz form per (c,ky): D[16x16] += A[16x24] (input rows) * B[24x16] (banded weights).
// B is precomputed per-lane in exact WMMA layout -> inner loop has zero predication.
__global__ void k_conv(const float* __restrict__ feat0, const float* __restrict__ feat1,
                       const float* __restrict__ conv_w, const float* __restrict__ params,
                       float* __restrict__ convo, double* __restrict__ bp2) {
  __shared__ float lds[2][22][137];  // rows y0-3..y0+18, cols x=-3..133 (137: bank-friendly)
  __shared__ v2f   btab[84][32];     // B operand per (c,ky,kk) in WMMA lane layout
  __shared__ double sh2[8][2];

  int b  = blockIdx.x >> 3;
  int y0 = (blockIdx.x & 7) * 16;
  int tid = threadIdx.x;

  float sc0 = params[0], bi0 = params[1], sc1 = params[2], bi1 = params[3];

  // stage banded weight matrices: btab[e=(c*7+ky)*6+kk][lane] = {B[k,ln], B[k+1,ln]}
  // where k = 4*kk + 2*half, B[k,n] = w[c][ky][k-n] for 0<=k-n<=6 else 0
  for (int i = tid; i < 84*32; i += 256) {
    int lane2 = i & 31;
    int e     = i >> 5;            // 0..83
    int kk    = e % 6;
    int cy    = e / 6;             // c*7 + ky
    int h = lane2 >> 4, l = lane2 & 15;
    int k = kk*4 + h*2;
    int d0 = k - l;
    v2f bv;
    bv.x = (d0   >= 0 && d0   <= 6) ? conv_w[cy*7 + d0]     : 0.f;
    bv.y = (d0+1 >= 0 && d0+1 <= 6) ? conv_w[cy*7 + d0 + 1] : 0.f;
    btab[e][lane2] = bv;
  }

  // stage normalized input strip (with halo + zero padding)
  for (int i = tid; i < 2*22*137; i += 256) {
    int c  = i / (22*137);
    int r  = i % (22*137);
    int yr = r / 137, xi = r % 137;
    int y = y0 + yr - 3, xx = xi - 3;
    float v = 0.f;
    if ((unsigned)y < HH && (unsigned)xx < WW) {
      float raw = (c ? feat1 : feat0)[(size_t)b*SPATIAL + y*WW + xx];
      v = c ? (raw*sc1 + bi1) : (raw*sc0 + bi0);
    }
    lds[c][yr][xi] = v;
  }
  __syncthreads();

  int wave = tid >> 5, lane = tid & 31;
  int half = lane >> 4, ln = lane & 15;
  int x0 = wave * 16;
  int ak = half * 2;                 // lane's K offset within a 4-block (A and B agree)

  v8f acc = {0.f,0.f,0.f,0.f,0.f,0.f,0.f,0.f};
  #pragma unroll
  for (int c = 0; c < 2; ++c) {
    #pragma unroll
    for (int ky = 0; ky < 7; ++ky) {
      const float* arow = &lds[c][ln + ky][x0 + ak];
      const v2f*   brow = &btab[(c*7 + ky)*6][lane];
      #pragma unroll
      for (int kk = 0; kk < 6; ++kk) {
        v2f a; a.x = arow[kk*4]; a.y = arow[kk*4 + 1];
        v2f bb = brow[kk*32];
        acc = __builtin_amdgcn_wmma_f32_16x16x4_f32(
                  false, a, false, bb, (short)0, acc, false, false);
      }
    }
  }

  // scatter D: vgpr r -> row (y0 + r + 8*half), col (x0 + ln)
  size_t base = (size_t)b*SPATIAL + (x0 + ln);
  double ls = 0.0, lq = 0.0;
  #pragma unroll
  for (int r = 0; r < 8; ++r) {
    float v = acc[r];
    convo[base + (size_t)(y0 + r + 8*half)*WW] = v;
    ls += v; lq += (double)v*v;
  }
  for (int off = 16; off > 0; off >>= 1) {
    ls += __shfl_down(ls, off, 32);
    lq += __shfl_down(lq, off, 32);
  }
  if (lane == 0) { sh2[wave][0] = ls; sh2[wave][1] = lq; }
  __syncthreads();
  if (tid == 0) {
    double r0=0, r1=0;
    for (int w = 0; w < 8; ++w) { r0 += sh2[w][0]; r1 += sh2[w][1]; }
    bp2[blockIdx.x*2+0] = r0; bp2[blockIdx.x*2+1] = r1;
  }
}

// ============ Kernel 4: finalize BN2 (256 partials) ============
__global__ void k_bn2(const double* __restrict__ bp2,
                      const float* __restrict__ gamma, const float* __restrict__ beta,
                      float* __restrict__ params) {
  __shared__ double sred[256];
  __shared__ double res[2];
  int t = threadIdx.x;
  for (int q = 0; q < 2; ++q) {
    sred[t] = bp2[t*2+q];
    __syncthreads();
    for (int s = 128; s > 0; s >>= 1) { if (t < s) sred[t] += sred[t+s]; __syncthreads(); }
    if (t == 0) res[q] = sred[0];
    __syncthreads();
  }
  if (t == 0) {
    double mean = res[0] / (double)NPOS;
    double var  = res[1] / (double)NPOS - mean*mean;
    double sc   = (double)gamma[0] / sqrt(var + 1e-5);
    params[4] = (float)sc;
    params[5] = (float)((double)beta[0] - mean*sc);
  }
}

// ============ Kernel 5: attention = sigmoid(bn2(conv)); out = att * x ============
// x is on last use, out is never re-read -> nontemporal to keep L2 clean.
__global__ void k_apply(const float* __restrict__ x, const float* __restrict__ convo,
                        const float* __restrict__ params, float* __restrict__ out) {
  int gid = blockIdx.x * 256 + threadIdx.x;
  int b   = gid / (SPATIAL/4);
  int hw4 = (gid % (SPATIAL/4)) * 4;
  float sc = params[4], bi = params[5];

  v4f t = *(const v4f*)(convo + (size_t)b*SPATIAL + hw4);
  v4f att;
  att.x = 1.f/(1.f + expf(-(t.x*sc + bi)));
  att.y = 1.f/(1.f + expf(-(t.y*sc + bi)));
  att.z = 1.f/(1.f + expf(-(t.z*sc + bi)));
  att.w = 1.f/(1.f + expf(-(t.w*sc + bi)));

  int c0 = blockIdx.y * 16;
  const v4f* xp = (const v4f*)(x   + ((size_t)b*CH + c0)*SPATIAL + hw4);
  v4f*       op = (v4f*)      (out + ((size_t)b*CH + c0)*SPATIAL + hw4);
  #pragma unroll 4
  for (int c = 0; c < 16; ++c) {
    if (c + 2 < 16)
      __builtin_prefetch((const char*)&xp[(c+2)*(SPATIAL/4)], 0, 0);
    v4f v = __builtin_nontemporal_load(&xp[c*(SPATIAL/4)]);
    v.x *= att.x; v.y *= att.y; v.z *= att.z; v.w *= att.w;
    __builtin_nontemporal_store(v, &op[c*(SPATIAL/4)]);
  }
}

extern "C" void kernel_launch(void* const* d_in, const int* in_sizes, int n_in,
                              void* d_out, int out_size, void* d_ws, size_t ws_size,
                              hipStream_t stream) {
  (void)in_sizes; (void)n_in; (void)out_size; (void)ws_size;
  const float* x      = (const float*)d_in[0];
  const float* conv_w = (const float*)d_in[1];
  const float* g1     = (const float*)d_in[2];
  const float* b1     = (const float*)d_in[3];
  const float* g2     = (const float*)d_in[4];
  const float* b2     = (const float*)d_in[5];
  float* out = (float*)d_out;

  char*   ws     = (char*)d_ws;
  float*  params = (float*)(ws + 64);
  double* bp1    = (double*)(ws + 1024);
  double* bp2    = (double*)(ws + 17408);
  float*  feat0  = (float*)(ws + 32768);
  float*  feat1  = feat0 + NPOS;
  float*  convo  = feat1 + NPOS;

  k_reduce<<<NPOS/4/256, 256, 0, stream>>>(x, feat0, feat1, bp1);
  k_bn1  <<<1,          256, 0, stream>>>(bp1, g1, b1, params);
  k_conv <<<BATCH*8,    256, 0, stream>>>(feat0, feat1, conv_w, params, convo, bp2);
  k_bn2  <<<1,          256, 0, stream>>>(bp2, g2, b2, params);
  dim3 g5(NPOS/4/256, 8, 1);
  k_apply<<<g5,         256, 0, stream>>>(x, convo, params, out);
}